// SimpleDepthRenderer_22565758173373
// MI455X (gfx1250) — compile-verified
//
#include <hip/hip_runtime.h>

typedef float v2f __attribute__((ext_vector_type(2)));
typedef float v8f __attribute__((ext_vector_type(8)));

#define IMG      256
#define MAX_R    64
#define N_AG     2047
#define N_OB     2048
#define N_ENT    (N_AG + N_OB)              /* 4095 */
#define HALF_FOV 0.78539816339744830962f    /* pi/4 (FOV_RAD/2) */
#define CAM_RANGE 3.0f

// ---------------------------------------------------------------------------
// Kernel 0: initialize image to 1.0f (harness poisons d_out; must re-init).
// ---------------------------------------------------------------------------
__global__ __launch_bounds__(256) void init_img_kernel(float* __restrict__ img) {
    int i = blockIdx.x * blockDim.x + threadIdx.x;
    if (i < IMG * IMG) img[i] = 1.0f;
}

// ---------------------------------------------------------------------------
// Kernel 1: per-entity transform. The rel @ rot.T product is computed with
// V_WMMA_F32_16X16X4_F32: 16 entities per wave.
//
//   A (16x4, docs-confirmed layout: lane m holds K0/K1; lane m+16 holds K2/K3):
//     row0 = ( c, 0, 0,  s)   -> cam_x = c*relx + s*rely
//     row1 = (-s, 0, 0,  c)   -> cam_y = -s*relx + c*rely
//   B (4x16): (relx, rely) replicated into BOTH lane halves. With the split
//     A rows above the dot product is correct regardless of whether the B
//     lane halves map to K rows (0,1)/(2,3) or (0,2)/(1,3).
//   D: lanes 0..15 get cam_x in VGPR0 (M=0 row) and cam_y in VGPR1 (M=1 row).
//
// WMMA executes in uniform control flow (EXEC all-ones); only the final
// store is lane-guarded.
// ---------------------------------------------------------------------------
__global__ __launch_bounds__(256) void transform_kernel(
    const float* __restrict__ agent_pos,
    const float* __restrict__ goal_pos,
    const float* __restrict__ other_agents,
    const float* __restrict__ obstacles,
    int4* __restrict__ params)
{
    const int lane  = threadIdx.x & 31;
    const int wave  = threadIdx.x >> 5;
    const int gwave = blockIdx.x * (blockDim.x >> 5) + wave;
    const int m     = lane & 15;
    const int ebase = gwave * 16;
    int  eidx     = ebase + m;
    const bool in_range = (eidx < N_ENT);
    if (!in_range) eidx = N_ENT - 1;   // clamp; no EXEC-losing early exit

    // Camera basis (uniform across wave; scalarized by the compiler).
    const float ax = agent_pos[0], ay = agent_pos[1];
    const float vx = goal_pos[0] - ax, vy = goal_pos[1] - ay;
    const float inv = 1.0f / (sqrtf(vx * vx + vy * vy) + 1e-8f);
    const float c = vy * inv, s = vx * inv;

    // Gather entity position / radius / height.
    float posx, posy, radius, height;
    if (eidx < N_AG) {
        posx = other_agents[2 * eidx];
        posy = other_agents[2 * eidx + 1];
        radius = 0.05f;   // AGENT_RADIUS
        height = 0.2f;    // AGENT_HEIGHT
    } else {
        const int o = eidx - N_AG;
        posx = obstacles[3 * o];
        posy = obstacles[3 * o + 1];
        radius = obstacles[3 * o + 2];
        height = 0.5f;    // OBSTACLE_HEIGHT
    }
    const float relx = posx - ax, rely = posy - ay;

    // Build WMMA operands (uniform flow; selects compile to v_cndmask).
    v2f a;
    a.x = (lane == 0) ? c : ((lane == 1) ? -s : 0.0f);   // K0 column
    a.y = (lane == 16) ? s : ((lane == 17) ? c : 0.0f);  // K3 column (lane-16 rows)
    v2f b;
    b.x = relx;
    b.y = rely;
    v8f acc = {};
    acc = __builtin_amdgcn_wmma_f32_16x16x4_f32(
        /*neg_a=*/false, a, /*neg_b=*/false, b,
        /*c_mod=*/(short)0, acc, /*reuse_a=*/false, /*reuse_b=*/false);

    const float cam_x = acc[0];  // M=0 row, lanes 0..15
    const float cam_y = acc[1];  // M=1 row, lanes 0..15

    const float dist  = sqrtf(cam_x * cam_x + cam_y * cam_y);
    const float angle = atan2f(cam_x, cam_y);  // jnp.arctan2(cam_x, cam_y)
    const bool valid = in_range && (cam_y >= 0.0f) && (dist <= CAM_RANGE) &&
                       (fabsf(angle) <= HALF_FOV);

    const float pixel_x = angle / HALF_FOV * 0.5f;
    const int px_int = (int)floorf((pixel_x + 0.5f) * (float)IMG);
    int pr = (int)floorf(radius / (dist + 1e-8f) * (float)IMG * 0.5f);
    pr = min(max(pr, 1), MAX_R);

    float depth = fminf(dist / CAM_RANGE, 1.0f) * (1.0f - height * 0.3f);
    depth = fmaxf(depth, 0.0f);

    if (lane < 16 && in_range) {
        int4 out;
        out.x = px_int;
        out.y = pr;
        out.z = __float_as_int(depth);
        out.w = valid ? 1 : 0;
        params[ebase + m] = out;
    }
}

// ---------------------------------------------------------------------------
// Kernel 2: one block per entity; threads stride the (2pr+1)^2 bounding box
// and min-splat the disk. Non-negative float bits are monotone as u32, so
// atomicMin on the bit pattern == exact float min (image starts at 1.0f).
// Masked-out / out-of-disk pixels contribute 1.0 in the reference == no-op.
// ---------------------------------------------------------------------------
__global__ __launch_bounds__(256) void scatter_kernel(
    const int4* __restrict__ params,
    float* __restrict__ img)
{
    const int4 p = params[blockIdx.x];  // uniform -> scalar load
    if (p.w == 0) return;
    const int pxc = p.x;
    const int pr  = p.y;
    const unsigned depth_bits = (unsigned)p.z;
    const int side  = 2 * pr + 1;
    const int total = side * side;

    for (int i = threadIdx.x; i < total; i += blockDim.x) {
        const int dy = i / side - pr;
        const int dx = i - (dy + pr) * side - pr;
        if (dx * dx + dy * dy <= pr * pr) {
            const int px = pxc + dx;
            const int py = (IMG / 2) + dy;
            if (px >= 0 && px < IMG && py >= 0 && py < IMG) {
                atomicMin((unsigned*)&img[py * IMG + px], depth_bits);
            }
        }
    }
}

// ---------------------------------------------------------------------------
extern "C" void kernel_launch(void* const* d_in, const int* in_sizes, int n_in,
                              void* d_out, int out_size, void* d_ws, size_t ws_size,
                              hipStream_t stream) {
    const float* agent_pos    = (const float*)d_in[0];
    const float* goal_pos     = (const float*)d_in[1];
    const float* other_agents = (const float*)d_in[2];
    const float* obstacles    = (const float*)d_in[3];
    float* img   = (float*)d_out;
    int4* params = (int4*)d_ws;   // 4095 * 16 B = 64 KB scratch

    init_img_kernel<<<(IMG * IMG + 255) / 256, 256, 0, stream>>>(img);
    // 32 blocks * 8 waves * 16 entities/wave = 4096 slots >= 4095 entities
    transform_kernel<<<32, 256, 0, stream>>>(agent_pos, goal_pos,
                                             other_agents, obstacles, params);
    scatter_kernel<<<N_ENT, 256, 0, stream>>>(params, img);
}